// Coefficients_36618891166258
// MI455X (gfx1250) — compile-verified
//
#include <hip/hip_runtime.h>
#include <hip/hip_bf16.h>
#include <stdint.h>

// Problem constants (from reference)
#define NN 2048            // N_NODES
#define NE 4096            // N_ELEMS
#define WROW (2 * NE + NN) // 10240 columns per row
#define Q4 (WROW / 4)      // 2560 float4 per full row

typedef float v4f __attribute__((ext_vector_type(4)));

__device__ __forceinline__ void st_nt4(float* p, v4f v) {
    __builtin_nontemporal_store(v, (v4f*)p);
}

// ---------------------------------------------------------------------------
// Rows [0, N): [ M | 0_{N,E} | 0_{N,N} ]
// One float4 per thread, full row width. N*Q4 = 2048*2560 = 5,242,880 threads.
// ---------------------------------------------------------------------------
__global__ __launch_bounds__(256) void fill_top(const float* __restrict__ M,
                                                float* __restrict__ out) {
    unsigned idx = blockIdx.x * 256u + threadIdx.x;
    if (idx >= (unsigned)(NN * Q4)) return;
    unsigned r  = idx / Q4;
    unsigned c4 = idx % Q4;
    v4f v = {0.f, 0.f, 0.f, 0.f};
    if (c4 < (unsigned)(NE / 4)) {
        v = ((const v4f*)M)[(size_t)r * (NE / 4) + c4];
    }
    st_nt4(out + (size_t)r * WROW + 4u * c4, v);
}

// ---------------------------------------------------------------------------
// Rows [N, N+E), cols [0, 2E): [ 0_{E,E} | I_E ]
// E * (2E/4) = 4096*2048 = 8,388,608 threads.
// ---------------------------------------------------------------------------
__global__ __launch_bounds__(256) void fill_mid(float* __restrict__ out) {
    unsigned idx = blockIdx.x * 256u + threadIdx.x;
    if (idx >= (unsigned)(NE * (2 * NE / 4))) return;
    unsigned i  = idx / (2 * NE / 4);
    unsigned c4 = idx % (2 * NE / 4);
    v4f v = {0.f, 0.f, 0.f, 0.f};
    unsigned d = (unsigned)NE + i; // identity diagonal column
    if ((d >> 2) == c4) v[d & 3u] = 1.0f;
    st_nt4(out + (size_t)(NN + i) * WROW + 4u * c4, v);
}

// ---------------------------------------------------------------------------
// Rows [N, N+E), cols [2E, 2E+N): -M^T
// 32x32 tiles; async global->LDS scatter performs the transpose in flight.
// Block (32,8): each wave32 is one (tx, fixed ty) row. grid (E/32, N/32).
// ---------------------------------------------------------------------------
__global__ __launch_bounds__(256) void transpose_neg(const float* __restrict__ M,
                                                     float* __restrict__ out) {
    __shared__ float lds[32 * 33]; // padded stride: 33 mod 64 banks -> conflict-free
    const unsigned tx = threadIdx.x; // 0..31
    const unsigned ty = threadIdx.y; // 0..7
    const unsigned i0 = blockIdx.x * 32u; // column tile of M  (E dim)
    const unsigned j0 = blockIdx.y * 32u; // row tile of M     (N dim)

    // Load M[j0+r][i0+tx] directly into lds[tx*33 + r] (transposed scatter),
    // via the CDNA5 async global->LDS path (tracked by ASYNCcnt).
#pragma unroll
    for (int k = 0; k < 4; ++k) {
        unsigned r = ty + 8u * k;
        const float* g = M + (size_t)(j0 + r) * NE + (i0 + tx);
        unsigned ldsoff = (unsigned)(size_t)&lds[tx * 33u + r]; // low 32b = LDS offset
        unsigned long long ga = (unsigned long long)(size_t)g;
        asm volatile("global_load_async_to_lds_b32 %0, %1, off"
                     :
                     : "v"(ldsoff), "v"(ga)
                     : "memory");
    }
    // Drain this wave's async ops, then sync the workgroup.
    asm volatile("s_wait_asynccnt 0" ::: "memory");
    __syncthreads();

    // Coalesced negated store: out[N+i0+r][2E+j0+tx] = -lds[r*33 + tx]
#pragma unroll
    for (int k = 0; k < 4; ++k) {
        unsigned r = ty + 8u * k;
        float v = -lds[r * 33u + tx];
        __builtin_nontemporal_store(
            v, out + (size_t)(NN + i0 + r) * WROW + (2u * NE + j0 + tx));
    }
}

// ---------------------------------------------------------------------------
// Rows [N+E, N+2E): diag(z_val) at col p, diag(y_val) at col E+p, zeros else.
// E * Q4 = 4096*2560 = 10,485,760 threads, full row width.
// ---------------------------------------------------------------------------
__global__ __launch_bounds__(256) void fill_bot(const float* __restrict__ params,
                                                const int* __restrict__ kinds,
                                                float* __restrict__ out) {
    unsigned idx = blockIdx.x * 256u + threadIdx.x;
    if (idx >= (unsigned)(NE * Q4)) return;
    unsigned p  = idx / Q4;
    unsigned c4 = idx % Q4;

    float pv = params[p];
    int   kv = kinds[p];
    // z: R -> -param ; VC or SW(off: param<=0) -> 1 ; else 0
    float z = (kv == 0) ? -pv
              : ((kv == 2 || (kv == 3 && !(pv > 0.f))) ? 1.f : 0.f);
    // y: R, IVS, or SW(on: param>0) -> 1 ; else 0
    float y = (kv == 0 || kv == 1 || (kv == 3 && pv > 0.f)) ? 1.f : 0.f;

    v4f v = {0.f, 0.f, 0.f, 0.f};
    if ((p >> 2) == c4) v[p & 3u] = z;
    unsigned d = (unsigned)NE + p;
    if ((d >> 2) == c4) v[d & 3u] = y;
    st_nt4(out + (size_t)(NN + NE + p) * WROW + 4u * c4, v);
}

// ---------------------------------------------------------------------------
extern "C" void kernel_launch(void* const* d_in, const int* in_sizes, int n_in,
                              void* d_out, int out_size, void* d_ws, size_t ws_size,
                              hipStream_t stream) {
    const float* M      = (const float*)d_in[0]; // (N, E) row-major
    const float* params = (const float*)d_in[1]; // (E,)
    const int*   kinds  = (const int*)d_in[2];   // (E,)
    float* out = (float*)d_out;                  // (2E+N, 2E+N)

    (void)in_sizes; (void)n_in; (void)out_size; (void)d_ws; (void)ws_size;

    // Each kernel writes a disjoint region; together they cover every element.
    fill_top<<<(NN * Q4) / 256, 256, 0, stream>>>(M, out);
    fill_mid<<<(NE * (2 * NE / 4)) / 256, 256, 0, stream>>>(out);
    transpose_neg<<<dim3(NE / 32, NN / 32), dim3(32, 8), 0, stream>>>(M, out);
    fill_bot<<<(NE * Q4) / 256, 256, 0, stream>>>(params, kinds, out);
}